// LorentzKG2_86337432584672
// MI455X (gfx1250) — compile-verified
//
#include <hip/hip_runtime.h>
#include <math.h>

typedef __attribute__((ext_vector_type(2))) float v2f;
typedef __attribute__((ext_vector_type(8))) float v8f;

#define RANK 33
#define KP   36      // K padded to multiple of 4 (zero-filled)
#define NENT 50000
#define NREL 500
#define NB   2000
#define NT   5       // N-tiles (of 16) per wave; 50000/16 = 3125 = 5*625

__device__ __forceinline__ float wred(float v) {
#pragma unroll
  for (int o = 16; o > 0; o >>= 1) v += __shfl_xor(v, o, 32);
  return v;
}

// ---------------- Kernel A: expmap0 over all entities -> embp (NENT x KP, padded) ----
__global__ void expmap_kernel(const float* __restrict__ ent, float* __restrict__ embp) {
  int row = blockIdx.x * blockDim.x + threadIdx.x;
  if (row >= NENT) return;
  const float* u = ent + (size_t)row * RANK;
  float ub[RANK];
  float s = 0.f;
#pragma unroll
  for (int i = 0; i < RANK; ++i) { float x = u[i]; ub[i] = x; s += x * x; }
  float en = sqrtf(s);
  float scale = fminf(1.f, 2.0f / fmaxf(en, 1e-12f));
#pragma unroll
  for (int i = 0; i < RANK; ++i) ub[i] *= scale;
  float inner = -ub[0] * ub[0];
#pragma unroll
  for (int i = 1; i < RANK; ++i) inner += ub[i] * ub[i];
  float nomin = sqrtf(fmaxf(inner, 1e-8f));
  float sh = sinhf(nomin) / nomin;
  float ssum = 0.f;
  float* o = embp + (size_t)row * KP;
#pragma unroll
  for (int i = 1; i < RANK; ++i) { float r = sh * ub[i]; ssum += r * r; o[i] = r; }
  o[0] = sqrtf(1.f + ssum);            // recomputed Lorentz time component
  o[33] = 0.f; o[34] = 0.f; o[35] = 0.f;
}

// ---------------- Kernel B: per-query Householder chain + boost -> lhsp (NB x KP) ----
__global__ void lhs_kernel(const float* __restrict__ embp, const float* __restrict__ rot,
                           const float* __restrict__ boosts, const int* __restrict__ q,
                           float* __restrict__ lhsp) {
  int wv = (blockIdx.x * blockDim.x + threadIdx.x) >> 5;
  int lane = threadIdx.x & 31;
  if (wv >= NB) return;
  int h = q[wv * 3 + 0];
  int r = q[wv * 3 + 1];
  float lt = embp[(size_t)h * KP];                 // time component
  float x  = embp[(size_t)h * KP + 1 + lane];      // space component (lane = i-1)
  const float* rr = rot + (size_t)r * 32 * 32;
#pragma unroll 4
  for (int w = 0; w < 32; ++w) {
    float g = rr[w * 32 + lane];
    float v = 0.5f * g * (1.f + erff(g * 0.70710678118654752f));  // exact GELU
    float n = wred(v * v);
    float d = wred(x * v);
    x -= 2.f * d * v / n;                          // x <- x H_w
  }
  float bj = tanhf(boosts[(size_t)r * 32 + lane]) * (1.f / 33.f);
  float b2 = wred(bj * bj);
  float bd = wred(bj * x);
  float zeta = 1.f / (sqrtf(1.f - b2) + 1e-8f);
  float x0 = zeta * lt - zeta * bd;
  float xr = -zeta * lt * bj + x + (zeta - 1.f) / (b2 + 1e-9f) * bj * bd;
  float* o = lhsp + (size_t)wv * KP;
  o[1 + lane] = xr;
  if (lane == 0) o[0] = -x0;                        // lhs_neg flips time sign
  if (lane < 3) o[33 + lane] = 0.f;                 // K padding
}

// ---------------- Kernel C: scores = 2 - 2 * lhs_neg @ emb^T via V_WMMA_F32_16X16X4_F32
__global__ void score_kernel(const float* __restrict__ lhsp, const float* __restrict__ embp,
                             float* __restrict__ out) {
  const int NGRP = (NENT / 16) / NT;                // 625 groups of NT tiles
  const int totalWaves = (NB / 16) * NGRP;          // 125 * 625 = 78125
  int wv = (blockIdx.x * blockDim.x + threadIdx.x) >> 5;
  int lane = threadIdx.x & 31;
  if (wv >= totalWaves) return;                     // wave-uniform: EXEC stays all-ones
  int mt = wv / NGRP;
  int ng = wv % NGRP;
  int m0 = mt * 16;
  int n0 = ng * (NT * 16);
  int mrow  = lane & 15;
  int khalf = (lane >> 4) * 2;                      // lanes 0-15: K {0,1}; 16-31: K {2,3}

  // Preload all 9 A fragments (16x4 f32 each): lane holds A[m0+mrow][4kk+khalf .. +1]
  v2f a[9];
#pragma unroll
  for (int kk = 0; kk < 9; ++kk) {
    const float* p = lhsp + (size_t)(m0 + mrow) * KP + 4 * kk + khalf;  // 8B aligned
    a[kk] = *(const v2f*)p;
  }

#pragma unroll
  for (int t = 0; t < NT; ++t) {
    int n = n0 + t * 16 + mrow;                     // B[K][N] = emb[n][K] (emb^T)
    const float* bp = embp + (size_t)n * KP + khalf;
    v8f c = {0.f, 0.f, 0.f, 0.f, 0.f, 0.f, 0.f, 0.f};
#pragma unroll
    for (int kk = 0; kk < 9; ++kk) {
      v2f b = *(const v2f*)(bp + 4 * kk);
      c = __builtin_amdgcn_wmma_f32_16x16x4_f32(false, a[kk], false, b,
                                                (short)0, c, false, false);
    }
#pragma unroll
    for (int r8 = 0; r8 < 8; ++r8) {
      int m = m0 + r8 + 8 * (lane >> 4);            // C layout: vgpr r -> M=r / M=r+8
      out[(size_t)m * NENT + (size_t)n] = 2.f - 2.f * c[r8];
    }
  }
}

// ---------------- Kernel D: per-relation Frobenius norms -------------------------------
__global__ void norm_kernel(const float* __restrict__ rot, const float* __restrict__ boosts,
                            float* __restrict__ rotn, float* __restrict__ bstn) {
  int wv = (blockIdx.x * blockDim.x + threadIdx.x) >> 5;
  int lane = threadIdx.x & 31;
  if (wv >= NREL) return;
  const float* rr = rot + (size_t)wv * 1024;
  float s = 0.f;
#pragma unroll
  for (int i = 0; i < 32; ++i) { float x = rr[i * 32 + lane]; s += x * x; }
  s = wred(s);
  float bx = boosts[(size_t)wv * 32 + lane];
  float bs = wred(bx * bx);
  if (lane == 0) { rotn[wv] = sqrtf(s); bstn[wv] = sqrtf(bs); }
}

// ---------------- Kernel E: reg[i][j] = bstn[i] + rotn[j]  (broadcast (500,)+(500,1)) ---
__global__ void reg_kernel(const float* __restrict__ rotn, const float* __restrict__ bstn,
                           float* __restrict__ out) {
  int idx = blockIdx.x * blockDim.x + threadIdx.x;
  if (idx >= NREL * NREL) return;
  int i = idx / NREL, j = idx % NREL;
  out[idx] = bstn[i] + rotn[j];
}

extern "C" void kernel_launch(void* const* d_in, const int* in_sizes, int n_in,
                              void* d_out, int out_size, void* d_ws, size_t ws_size,
                              hipStream_t stream) {
  const float* ent    = (const float*)d_in[0];
  const float* rot    = (const float*)d_in[1];
  const float* boosts = (const float*)d_in[2];
  const int*   q      = (const int*)d_in[3];
  float* out = (float*)d_out;

  // workspace layout (floats): embp[NENT*KP] | lhsp[NB*KP] | rotn[NREL] | bstn[NREL]
  float* embp = (float*)d_ws;
  float* lhsp = embp + (size_t)NENT * KP;
  float* rotn = lhsp + (size_t)NB * KP;
  float* bstn = rotn + NREL;

  hipLaunchKernelGGL(expmap_kernel, dim3((NENT + 255) / 256), dim3(256), 0, stream,
                     ent, embp);
  hipLaunchKernelGGL(lhs_kernel, dim3((NB * 32 + 255) / 256), dim3(256), 0, stream,
                     embp, rot, boosts, q, lhsp);
  hipLaunchKernelGGL(norm_kernel, dim3((NREL * 32 + 255) / 256), dim3(256), 0, stream,
                     rot, boosts, rotn, bstn);

  const int totalWaves = (NB / 16) * ((NENT / 16) / NT);   // 78125
  hipLaunchKernelGGL(score_kernel,
                     dim3((totalWaves * 32 + 255) / 256), dim3(256), 0, stream,
                     lhsp, embp, out);
  hipLaunchKernelGGL(reg_kernel, dim3((NREL * NREL + 255) / 256), dim3(256), 0, stream,
                     rotn, bstn, out + (size_t)NB * NENT);
}